// ControlValLoss_5042291605607
// MI455X (gfx1250) — compile-verified
//
#include <hip/hip_runtime.h>
#include <stdint.h>

// Problem constants (fixed by the reference harness)
#define B_        64
#define NTRIP     2048
#define T_        6146            // 3*2048 + 2
#define TOKN      204
#define REVSPLIT  101
#define PADIDX    203
#define NTOTAL    (B_ * NTRIP)    // 131072 triples
#define BLOCKS    2048
#define WAVES_PB  8
#define TOTWAVES  (BLOCKS * WAVES_PB)   // 16384
#define TPW       (NTOTAL / TOTWAVES)   // 8 triples per wave
#define ROWB      (TOKN * 4)            // 816 bytes per row
#define TRIB      (3 * ROWB)            // 2448 bytes per triple

// CDNA5 async global->LDS copy (ASYNCcnt-tracked); lds_addr is the 32-bit LDS
// byte offset (low half of the generic shared pointer), gptr the per-lane
// global source.  Each enabled lane moves 16 bytes.
__device__ __forceinline__ void async_ld_b128(uint32_t lds_addr, const float* gptr) {
  asm volatile("global_load_async_to_lds_b128 %0, %1, off"
               :: "v"(lds_addr), "v"(gptr) : "memory");
}

// Wave32 argmax over one 204-float row resident in LDS (first-occurrence wins).
__device__ __forceinline__ int row_argmax(const float* row, int lane) {
  const float4* r4 = (const float4*)row;
  float4 a = r4[lane];
  int base = lane << 2;
  float bv = a.x; int bi = base;
  if (a.y > bv) { bv = a.y; bi = base + 1; }
  if (a.z > bv) { bv = a.z; bi = base + 2; }
  if (a.w > bv) { bv = a.w; bi = base + 3; }
  if (lane < 19) {                       // tail chunk: elements 128..203
    float4 b = r4[32 + lane];
    int b2 = 128 + (lane << 2);
    if (b.x > bv) { bv = b.x; bi = b2; }
    if (b.y > bv) { bv = b.y; bi = b2 + 1; }
    if (b.z > bv) { bv = b.z; bi = b2 + 2; }
    if (b.w > bv) { bv = b.w; bi = b2 + 3; }
  }
#pragma unroll
  for (int off = 16; off; off >>= 1) {
    float ov = __shfl_xor(bv, off, 32);
    int   oi = __shfl_xor(bi, off, 32);
    if (ov > bv || (ov == bv && oi < bi)) { bv = ov; bi = oi; }
  }
  return bi;
}

// Softmax mass split at class REVSPLIT for one 204-float row in LDS.
__device__ __forceinline__ void row_softsplit(const float* row, int lane,
                                              float& s_no, float& s_all) {
  const float4* r4 = (const float4*)row;
  float4 a = r4[lane];
  float4 b = make_float4(-__builtin_inff(), -__builtin_inff(),
                         -__builtin_inff(), -__builtin_inff());
  if (lane < 19) b = r4[32 + lane];
  float lm = fmaxf(fmaxf(a.x, a.y), fmaxf(a.z, a.w));
  lm = fmaxf(lm, fmaxf(fmaxf(b.x, b.y), fmaxf(b.z, b.w)));
#pragma unroll
  for (int off = 16; off; off >>= 1) lm = fmaxf(lm, __shfl_xor(lm, off, 32));

  int base = lane << 2;
  float sa = 0.f, sn = 0.f, e;
  e = __expf(a.x - lm); sa += e; if (base + 0 < REVSPLIT) sn += e;
  e = __expf(a.y - lm); sa += e; if (base + 1 < REVSPLIT) sn += e;
  e = __expf(a.z - lm); sa += e; if (base + 2 < REVSPLIT) sn += e;
  e = __expf(a.w - lm); sa += e; if (base + 3 < REVSPLIT) sn += e;
  if (lane < 19) {                       // classes 128.. are all on the "yes" side
    sa += __expf(b.x - lm) + __expf(b.y - lm) +
          __expf(b.z - lm) + __expf(b.w - lm);
  }
#pragma unroll
  for (int off = 16; off; off >>= 1) {
    sa += __shfl_xor(sa, off, 32);
    sn += __shfl_xor(sn, off, 32);
  }
  s_no = sn; s_all = sa;
}

__global__ __launch_bounds__(256)
void cv_loss_main(const float* __restrict__ pred,
                  const float* __restrict__ gt_acc,
                  const float* __restrict__ gt_steer,
                  const int*   __restrict__ gt_rev,
                  float* __restrict__ part) {
  __shared__ __align__(16) float sbuf[WAVES_PB][2][3 * TOKN];  // double-buffered triples
  __shared__ float wpart[WAVES_PB][4];

  const int tid  = threadIdx.x;
  const int w    = tid >> 5;
  const int lane = tid & 31;
  const int wgl  = blockIdx.x * WAVES_PB + w;        // global wave id, 0..16383

  const uint32_t lds_base = (uint32_t)(uintptr_t)(&sbuf[w][0][0]);

  // Issue the 6 async b128 copies for one triple into the given stage buffer.
  auto issue = [&](int g, int stage) {
    const int b = g >> 11;
    const int j = g & (NTRIP - 1);
    const float* rowp = pred + (size_t)(b * T_ + 3 * j) * TOKN;
    const uint32_t lbase = lds_base + (uint32_t)stage * TRIB;
#pragma unroll
    for (int r = 0; r < 3; ++r) {
      const float* rp = rowp + r * TOKN;
      async_ld_b128(lbase + r * ROWB + lane * 16, rp + lane * 4);
      if (lane < 19)
        async_ld_b128(lbase + r * ROWB + 512 + lane * 16, rp + 128 + lane * 4);
    }
  };

  float accL = 0.f, steerL = 0.f, nllL = 0.f, vcnt = 0.f;

  issue(wgl, 0);                                     // prime the pipeline

#pragma unroll 1
  for (int s = 0; s < TPW; ++s) {
    const int g     = wgl + (s << 14);               // stride-16384 triple ids
    const int stage = s & 1;
    if (s + 1 < TPW) {
      issue(wgl + ((s + 1) << 14), stage ^ 1);       // prefetch next triple
      asm volatile("s_wait_asynccnt 0x6" ::: "memory");  // older 6 ops done
    } else {
      asm volatile("s_wait_asynccnt 0x0" ::: "memory");
    }

    const float* lrow = &sbuf[w][stage][0];
    const int aidx = row_argmax(lrow, lane);
    const int sidx = row_argmax(lrow + TOKN, lane);
    float s_no, s_all;
    row_softsplit(lrow + 2 * TOKN, lane, s_no, s_all);

    // acc: pred_acc = |tok/100 - 1|, smooth-L1 vs gt_acc
    float pa = fabsf((float)aidx / 100.0f - 1.0f);
    float d  = pa - gt_acc[g];
    float ad = fabsf(d);
    accL += (ad < 1.0f) ? 0.5f * d * d : ad - 0.5f;

    // steer: pred_steer = tok/100 - 1
    float ps = (float)sidx / 100.0f - 1.0f;
    d  = ps - gt_steer[g];
    ad = fabsf(d);
    steerL += (ad < 1.0f) ? 0.5f * d * d : ad - 0.5f;

    // reverse: probabilities used as 2-way logits
    float pno  = s_no / s_all;
    float pyes = (s_all - s_no) / s_all;
    int   gr   = gt_rev[g];
    float mm   = fmaxf(pno, pyes);
    float lse  = mm + __logf(__expf(pno - mm) + __expf(pyes - mm));
    if (gr != PADIDX) {
      nllL += lse - ((gr == 1) ? pyes : pno);
      vcnt += 1.0f;
    }
  }

  if (lane == 0) {
    wpart[w][0] = accL; wpart[w][1] = steerL;
    wpart[w][2] = nllL; wpart[w][3] = vcnt;
  }
  __syncthreads();
  if (tid == 0) {
    float a = 0.f, st = 0.f, n = 0.f, v = 0.f;
#pragma unroll
    for (int i = 0; i < WAVES_PB; ++i) {
      a += wpart[i][0]; st += wpart[i][1]; n += wpart[i][2]; v += wpart[i][3];
    }
    part[blockIdx.x * 4 + 0] = a;
    part[blockIdx.x * 4 + 1] = st;
    part[blockIdx.x * 4 + 2] = n;
    part[blockIdx.x * 4 + 3] = v;
  }
}

__global__ __launch_bounds__(256)
void cv_loss_final(const float* __restrict__ part, float* __restrict__ out) {
  __shared__ float red[256][4];
  const int tid = threadIdx.x;
  float a = 0.f, st = 0.f, n = 0.f, v = 0.f;
  for (int i = tid; i < BLOCKS; i += 256) {          // fixed order -> deterministic
    a  += part[4 * i + 0];
    st += part[4 * i + 1];
    n  += part[4 * i + 2];
    v  += part[4 * i + 3];
  }
  red[tid][0] = a; red[tid][1] = st; red[tid][2] = n; red[tid][3] = v;
  __syncthreads();
  for (int off = 128; off; off >>= 1) {
    if (tid < off) {
      red[tid][0] += red[tid + off][0];
      red[tid][1] += red[tid + off][1];
      red[tid][2] += red[tid + off][2];
      red[tid][3] += red[tid + off][3];
    }
    __syncthreads();
  }
  if (tid == 0) {
    out[0] = (red[0][0] + red[0][1]) * (1.0f / (float)NTOTAL);  // mean + mean
    out[1] = red[0][2] / fmaxf(red[0][3], 1.0f);
  }
}

extern "C" void kernel_launch(void* const* d_in, const int* in_sizes, int n_in,
                              void* d_out, int out_size, void* d_ws, size_t ws_size,
                              hipStream_t stream) {
  (void)in_sizes; (void)n_in; (void)out_size; (void)ws_size;
  const float* pred     = (const float*)d_in[0];
  const float* gt_acc   = (const float*)d_in[1];
  const float* gt_steer = (const float*)d_in[2];
  const int*   gt_rev   = (const int*)d_in[3];
  float* out  = (float*)d_out;
  float* part = (float*)d_ws;                        // 2048 * 4 floats = 32 KB

  cv_loss_main<<<BLOCKS, 256, 0, stream>>>(pred, gt_acc, gt_steer, gt_rev, part);
  cv_loss_final<<<1, 256, 0, stream>>>(part, out);
}